// DeltaLagModel_45801531244604
// MI455X (gfx1250) — compile-verified
//
#include <hip/hip_runtime.h>
#include <hip/hip_bf16.h>

// Shapes
#define BATCH   512
#define SEQL    30
#define FEAT    64
#define HID     64
#define G4      256      // 4*HID
#define NLEAD   500
#define MAXLAG  10
#define ROWS    5000     // NLEAD*MAXLAG
#define TOPK    5
#define INV_SCALE 0.125f // 1/sqrt(64)

typedef float v2f __attribute__((ext_vector_type(2)));
typedef float v8f __attribute__((ext_vector_type(8)));

static __device__ __forceinline__ v8f wmma_f32(v2f a, v2f b, v8f c) {
  // D = A(16x4 f32) x B(4x16 f32) + C(16x16 f32)
  return __builtin_amdgcn_wmma_f32_16x16x4_f32(
      /*neg_a=*/false, a, /*neg_b=*/false, b,
      /*c_mod=*/(short)0, c, /*reuse_a=*/false, /*reuse_b=*/false);
}

static __device__ __forceinline__ float sigf(float x) {
  return 1.0f / (1.0f + __expf(-x));
}

// ---------------------------------------------------------------------------
// Kernel 1: LSTM over time, 16 batch rows per block, 16 waves (512 thr).
// Each wave owns one 16-wide column tile of the 256-wide gate matrix.
// Weights (B operands) live in registers; x_t / h / c / gates live in LDS.
// fp32 WMMA 16x16x4 keeps exact fp32 semantics.
// ---------------------------------------------------------------------------
__global__ __launch_bounds__(512) void lstm_kernel(
    const float* __restrict__ x,      // [B,L,F]
    const float* __restrict__ W_ih,   // [4H,F]
    const float* __restrict__ W_hh,   // [4H,H]
    const float* __restrict__ b_ih,   // [4H]
    const float* __restrict__ b_hh,   // [4H]
    float* __restrict__ h_out) {      // [B,H]
  __shared__ float sX[16 * FEAT];
  __shared__ float sH[16 * HID];
  __shared__ float sC[16 * HID];
  __shared__ float sG[16 * G4];

  const int tid  = threadIdx.x;
  const int lane = tid & 31;
  const int wv   = tid >> 5;          // 0..15  -> column tile
  const int ln   = lane & 15;         // row/col within tile
  const int hi   = lane >> 4;         // 0 or 1 (K sub-pair / M half for D)
  const int kb   = hi * 2;            // K offset for A/B operands
  const int n0   = wv * 16;
  const int col  = n0 + ln;           // gate column this lane produces
  const int b0   = blockIdx.x * 16;

  // Hoist B operands (weight slices) into registers: 16 K-slices per matrix.
  v2f Bih[16], Bhh[16];
#pragma unroll
  for (int s = 0; s < 16; ++s) {
    const int k0 = s * 4 + kb;
    Bih[s].x = W_ih[col * FEAT + k0];
    Bih[s].y = W_ih[col * FEAT + k0 + 1];
    Bhh[s].x = W_hh[col * HID + k0];
    Bhh[s].y = W_hh[col * HID + k0 + 1];
  }
  const float bias = b_ih[col] + b_hh[col];

  // zero-init h, c
  for (int id = tid; id < 16 * HID; id += 512) { sH[id] = 0.0f; sC[id] = 0.0f; }
  __syncthreads();

  for (int t = 0; t < SEQL; ++t) {
    // stage x_t tile
    for (int id = tid; id < 16 * FEAT; id += 512) {
      const int m = id >> 6, f = id & 63;
      sX[id] = x[(size_t)(b0 + m) * (SEQL * FEAT) + t * FEAT + f];
    }
    __syncthreads();

    v8f acc;
#pragma unroll
    for (int r = 0; r < 8; ++r) acc[r] = bias;

#pragma unroll
    for (int s = 0; s < 16; ++s) {          // x_t @ W_ih^T  (K = 64)
      const int k0 = s * 4 + kb;
      v2f a; a.x = sX[ln * FEAT + k0]; a.y = sX[ln * FEAT + k0 + 1];
      acc = wmma_f32(a, Bih[s], acc);
    }
#pragma unroll
    for (int s = 0; s < 16; ++s) {          // h @ W_hh^T    (K = 64)
      const int k0 = s * 4 + kb;
      v2f a; a.x = sH[ln * HID + k0]; a.y = sH[ln * HID + k0 + 1];
      acc = wmma_f32(a, Bhh[s], acc);
    }

    // D layout: VGPR r -> row (r + 8*hi), col ln
#pragma unroll
    for (int r = 0; r < 8; ++r) sG[(r + 8 * hi) * G4 + col] = acc[r];
    __syncthreads();

    // gate nonlinearities + state update (torch order i,f,g,o)
    for (int id = tid; id < 16 * HID; id += 512) {
      const int m = id >> 6, hh = id & 63;
      const float gi = sigf(sG[m * G4 + hh]);
      const float gf = sigf(sG[m * G4 + 64 + hh]);
      const float gg = tanhf(sG[m * G4 + 128 + hh]);
      const float go = sigf(sG[m * G4 + 192 + hh]);
      const float c  = gf * sC[id] + gi * gg;
      sC[id] = c;
      sH[id] = go * tanhf(c);
    }
    __syncthreads();
  }

  for (int id = tid; id < 16 * HID; id += 512) {
    const int m = id >> 6, hh = id & 63;
    h_out[(size_t)(b0 + m) * HID + hh] = sH[id];
  }
}

// ---------------------------------------------------------------------------
// Kernel 2: qk = ((h_last @ W_Q^T) @ W_K) / sqrt(H)   [512,64] fp32 WMMA
// One wave per 16 batch rows.
// ---------------------------------------------------------------------------
__global__ __launch_bounds__(32) void qk_kernel(
    const float* __restrict__ h_last, // [B,H]
    const float* __restrict__ W_Q,    // [H,H]
    const float* __restrict__ W_K,    // [H,F]
    float* __restrict__ qk) {         // [B,F]
  __shared__ float sQ[16 * HID];
  const int lane = threadIdx.x & 31;
  const int ln = lane & 15;
  const int hi = lane >> 4;
  const int kb = hi * 2;
  const int b0 = blockIdx.x * 16;

  // query = h_last @ W_Q^T
#pragma unroll
  for (int nt = 0; nt < 4; ++nt) {
    const int n0 = nt * 16;
    v8f acc = {};
#pragma unroll
    for (int s = 0; s < 16; ++s) {
      const int k0 = s * 4 + kb;
      v2f a; a.x = h_last[(size_t)(b0 + ln) * HID + k0];
             a.y = h_last[(size_t)(b0 + ln) * HID + k0 + 1];
      v2f w; w.x = W_Q[(n0 + ln) * HID + k0];
             w.y = W_Q[(n0 + ln) * HID + k0 + 1];
      acc = wmma_f32(a, w, acc);
    }
#pragma unroll
    for (int r = 0; r < 8; ++r) sQ[(r + 8 * hi) * HID + n0 + ln] = acc[r];
  }
  __syncthreads();

  // qk = query @ W_K, fold in 1/sqrt(H)
#pragma unroll
  for (int nt = 0; nt < 4; ++nt) {
    const int n0 = nt * 16;
    v8f acc = {};
#pragma unroll
    for (int s = 0; s < 16; ++s) {
      const int k0 = s * 4 + kb;
      v2f a; a.x = sQ[ln * HID + k0]; a.y = sQ[ln * HID + k0 + 1];
      v2f w; w.x = W_K[k0 * FEAT + n0 + ln];
             w.y = W_K[(k0 + 1) * FEAT + n0 + ln];
      acc = wmma_f32(a, w, acc);
    }
#pragma unroll
    for (int r = 0; r < 8; ++r)
      qk[(size_t)(b0 + r + 8 * hi) * FEAT + n0 + ln] = acc[r] * INV_SCALE;
  }
}

// ---------------------------------------------------------------------------
// Kernel 3: attn[b,m] = raw[b,m,:] . qk[b,:]   (655 MB stream, HBM-bound)
// 16-lane groups: each group owns one row, lane ln holds float4 chunk ln.
// Fully coalesced 256B per half-wave iteration.
// ---------------------------------------------------------------------------
__global__ __launch_bounds__(256) void attn_kernel(
    const float* __restrict__ raw,  // [B,ROWS,F]
    const float* __restrict__ qk,   // [B,F] (already scaled)
    float* __restrict__ attn) {     // [B,ROWS]
  const int b   = blockIdx.y;
  const int tid = threadIdx.x;
  const int grp = tid >> 4;   // 0..15 groups per block
  const int ln  = tid & 15;

  const float4 q = ((const float4*)(qk + (size_t)b * FEAT))[ln];
  const int rowsPer = (ROWS + gridDim.x - 1) / gridDim.x;
  const int r0 = blockIdx.x * rowsPer;
  const int r1 = min(r0 + rowsPer, ROWS);
  const float4* base = (const float4*)(raw + (size_t)b * ROWS * FEAT);

  for (int r = r0 + grp; r < r1; r += 16) {
    const float4 v = base[(size_t)r * 16 + ln];
    float p = v.x * q.x + v.y * q.y + v.z * q.z + v.w * q.w;
    p += __shfl_xor(p, 8);
    p += __shfl_xor(p, 4);
    p += __shfl_xor(p, 2);
    p += __shfl_xor(p, 1);
    if (ln == 0) attn[(size_t)b * ROWS + r] = p;
  }
}

// ---------------------------------------------------------------------------
// Kernel 4: per-batch top-5 + softmax + gather + MLP. One block per batch row.
// ---------------------------------------------------------------------------
__global__ __launch_bounds__(256) void topk_mlp_kernel(
    const float* __restrict__ attn,  // [B,ROWS]
    const float* __restrict__ raw,   // [B,ROWS,F]
    const float* __restrict__ Wp1, const float* __restrict__ bp1,
    const float* __restrict__ Wp2, const float* __restrict__ bp2,
    const float* __restrict__ Wp3, const float* __restrict__ bp3,
    float* __restrict__ pred,        // [B]
    int*   __restrict__ idxOut,      // [B,5,2]
    float* __restrict__ scoreOut) {  // [B,5]
  const int b = blockIdx.x;
  const int tid = threadIdx.x;
  const float* arow = attn + (size_t)b * ROWS;

  const float NEG = -3.0e38f;
  float tv[TOPK]; int ti[TOPK];
#pragma unroll
  for (int k = 0; k < TOPK; ++k) { tv[k] = NEG; ti[k] = 0x7fffffff; }

  for (int i = tid; i < ROWS; i += 256) {
    const float v = arow[i];
    if (v > tv[TOPK - 1] || (v == tv[TOPK - 1] && i < ti[TOPK - 1])) {
      tv[TOPK - 1] = v; ti[TOPK - 1] = i;
#pragma unroll
      for (int k = TOPK - 1; k > 0; --k) {
        const bool sw = (tv[k] > tv[k - 1]) ||
                        (tv[k] == tv[k - 1] && ti[k] < ti[k - 1]);
        if (sw) {
          float fv = tv[k]; tv[k] = tv[k - 1]; tv[k - 1] = fv;
          int iv = ti[k]; ti[k] = ti[k - 1]; ti[k - 1] = iv;
        }
      }
    }
  }

  __shared__ float cv[256 * TOPK];
  __shared__ int   ci[256 * TOPK];
#pragma unroll
  for (int k = 0; k < TOPK; ++k) { cv[tid * TOPK + k] = tv[k]; ci[tid * TOPK + k] = ti[k]; }
  __syncthreads();

  __shared__ float selS[TOPK];
  __shared__ int   selI[TOPK];
  __shared__ float wsum[FEAT];
  __shared__ float h1v[HID];
  __shared__ float h2v[HID / 2];

  if (tid == 0) {
    float selV[TOPK];
    for (int k = 0; k < TOPK; ++k) {
      float bv = NEG; int bi = 0x7fffffff; int bj = 0;
      for (int j = 0; j < 256 * TOPK; ++j) {
        const float v = cv[j]; const int ix = ci[j];
        if (v > bv || (v == bv && ix < bi)) { bv = v; bi = ix; bj = j; }
      }
      selV[k] = bv; selI[k] = bi;
      cv[bj] = NEG; ci[bj] = 0x7fffffff;
    }
    // softmax over the 5 (already descending) values
    const float mx = selV[0];
    float s = 0.0f; float ex[TOPK];
    for (int k = 0; k < TOPK; ++k) { ex[k] = __expf(selV[k] - mx); s += ex[k]; }
    const float inv = 1.0f / s;
    for (int k = 0; k < TOPK; ++k) {
      selS[k] = ex[k] * inv;
      scoreOut[(size_t)b * TOPK + k] = selS[k];
      idxOut[((size_t)b * TOPK + k) * 2 + 0] = selI[k] / MAXLAG;
      idxOut[((size_t)b * TOPK + k) * 2 + 1] = selI[k] % MAXLAG;
    }
  }
  __syncthreads();

  if (tid < FEAT) {
    float acc = 0.0f;
#pragma unroll
    for (int k = 0; k < TOPK; ++k)
      acc += selS[k] * raw[(size_t)b * ROWS * FEAT + (size_t)selI[k] * FEAT + tid];
    wsum[tid] = acc;
  }
  __syncthreads();

  if (tid < HID) {
    float acc = bp1[tid];
    for (int f = 0; f < FEAT; ++f) acc += Wp1[tid * FEAT + f] * wsum[f];
    h1v[tid] = fmaxf(acc, 0.0f);
  }
  __syncthreads();

  if (tid < HID / 2) {
    float acc = bp2[tid];
    for (int f = 0; f < HID; ++f) acc += Wp2[tid * HID + f] * h1v[f];
    h2v[tid] = fmaxf(acc, 0.0f);
  }
  __syncthreads();

  if (tid == 0) {
    float acc = bp3[0];
    for (int j = 0; j < HID / 2; ++j) acc += Wp3[j] * h2v[j];
    pred[b] = acc;
  }
}

// ---------------------------------------------------------------------------
extern "C" void kernel_launch(void* const* d_in, const int* in_sizes, int n_in,
                              void* d_out, int out_size, void* d_ws, size_t ws_size,
                              hipStream_t stream) {
  const float* target = (const float*)d_in[0];
  const float* raw    = (const float*)d_in[1];
  const float* W_ih   = (const float*)d_in[2];
  const float* W_hh   = (const float*)d_in[3];
  const float* b_ih   = (const float*)d_in[4];
  const float* b_hh   = (const float*)d_in[5];
  const float* W_Q    = (const float*)d_in[6];
  const float* W_K    = (const float*)d_in[7];
  const float* Wp1    = (const float*)d_in[8];
  const float* bp1    = (const float*)d_in[9];
  const float* Wp2    = (const float*)d_in[10];
  const float* bp2    = (const float*)d_in[11];
  const float* Wp3    = (const float*)d_in[12];
  const float* bp3    = (const float*)d_in[13];

  // Output layout (flat, return order): pred[512] | idx[512*5*2 i32] |
  // scores[512*5] | attn[512*5000]
  float* out      = (float*)d_out;
  float* pred     = out;
  int*   idxOut   = (int*)(out + BATCH);
  float* scoreOut = out + BATCH + BATCH * TOPK * 2;
  float* attnOut  = out + BATCH + BATCH * TOPK * 2 + BATCH * TOPK;

  // Workspace: h_last[512*64] | qk[512*64]
  float* h_last = (float*)d_ws;
  float* qk     = h_last + BATCH * HID;

  lstm_kernel<<<BATCH / 16, 512, 0, stream>>>(target, W_ih, W_hh, b_ih, b_hh, h_last);
  qk_kernel<<<BATCH / 16, 32, 0, stream>>>(h_last, W_Q, W_K, qk);
  attn_kernel<<<dim3(20, BATCH), 256, 0, stream>>>(raw, qk, attnOut);
  topk_mlp_kernel<<<BATCH, 256, 0, stream>>>(attnOut, raw, Wp1, bp1, Wp2, bp2,
                                             Wp3, bp3, pred, idxOut, scoreOut);
}